// QuanvLayer_17325898072362
// MI455X (gfx1250) — compile-verified
//
#include <hip/hip_runtime.h>
#include <math.h>

typedef __attribute__((ext_vector_type(2))) float v2f;
typedef __attribute__((ext_vector_type(8))) float v8f;

#define NBASIS 81
#define NCHUNK 21           // K padded 81 -> 84, 4 per WMMA step
#define KPAD   (NCHUNK * 4)

// ------------------------------------------------------------------
// Setup: build coefficient matrix Cmat[KPAD][16] (cols 0..3 valid)
// out[p,w] = sum_m Phi[p,m] * Cmat[m][w],
// Phi[p,m] = prod_u f_{m_u}(theta_u), f = {1, cos, sin}, m base-3 (wire0 = most significant digit)
// ------------------------------------------------------------------
__global__ void quanv_setup(const float* __restrict__ weights, float* __restrict__ Cmat)
{
    __shared__ float Gre[256], Gim[256];
    __shared__ float Mre[4][256], Mim[4][256];
    const int t = threadIdx.x;

    // zero padded coefficient matrix
    for (int i = t; i < KPAD * 16; i += blockDim.x) Cmat[i] = 0.0f;

    // Rot(phi,theta,omega) = [[a,b],[c,d]] per wire (redundant per thread, cheap)
    float Rr[4][2][2], Ri[4][2][2];
    for (int w = 0; w < 4; ++w) {
        float phi = weights[w * 3 + 0], th = weights[w * 3 + 1], om = weights[w * 3 + 2];
        float ct = cosf(0.5f * th), st = sinf(0.5f * th);
        float apo = 0.5f * (phi + om), amo = 0.5f * (phi - om);
        Rr[w][0][0] =  cosf(apo) * ct;  Ri[w][0][0] = -sinf(apo) * ct;   // a = e^{-i(phi+om)/2} c
        Rr[w][0][1] = -cosf(amo) * st;  Ri[w][0][1] = -sinf(amo) * st;   // b = -e^{+i(phi-om)/2} s
        Rr[w][1][0] =  cosf(amo) * st;  Ri[w][1][0] = -sinf(amo) * st;   // c = e^{-i(phi-om)/2} s
        Rr[w][1][1] =  cosf(apo) * ct;  Ri[w][1][1] =  sinf(apo) * ct;   // d = e^{+i(phi+om)/2} c
    }

    // G[k][j] = A[sigma(k)][j], A = R0 (x) R1 (x) R2 (x) R3 (wire0 = MSB of index),
    // sigma = cnot01( cnot12( cnot23(k) ) ); wire w <-> bit (3-w)
    if (t < 256) {
        int k = t >> 4, j = t & 15;
        int r = k;
        r ^= ((r >> 1) & 1) << 0;   // CNOT(2,3)
        r ^= ((r >> 2) & 1) << 1;   // CNOT(1,2)
        r ^= ((r >> 3) & 1) << 2;   // CNOT(0,1)
        float are = 1.f, aim = 0.f;
        for (int w = 0; w < 4; ++w) {
            int bi = (r >> (3 - w)) & 1, bj = (j >> (3 - w)) & 1;
            float br = Rr[w][bi][bj], bm = Ri[w][bi][bj];
            float nr = are * br - aim * bm;
            float ni = are * bm + aim * br;
            are = nr; aim = ni;
        }
        Gre[(k << 4) | j] = are;  Gim[(k << 4) | j] = aim;
    }
    __syncthreads();

    // M_w = G^dagger S_w G  (S_w = diag(+1 if wire-w bit 0 else -1))
    if (t < 256) {
        int i = t >> 4, j = t & 15;
        for (int w = 0; w < 4; ++w) {
            float mr = 0.f, mi = 0.f;
            for (int k = 0; k < 16; ++k) {
                float sgn = ((k >> (3 - w)) & 1) ? -1.f : 1.f;
                float gr = Gre[(k << 4) | i], gi = Gim[(k << 4) | i];
                float hr = Gre[(k << 4) | j], hi = Gim[(k << 4) | j];
                mr += sgn * (gr * hr + gi * hi);   // conj(G[k,i]) * G[k,j]
                mi += sgn * (gr * hi - gi * hr);
            }
            Mre[w][t] = mr;  Mim[w][t] = mi;
        }
    }
    __syncthreads();

    // coef_w[m] = Re( sum_{i,j} M_w[i,j] * prod_u gamma_{m_u}(bi_u, bj_u) )
    //  gamma_1(=const): diag 1/2 ; gamma_cos: diag(+1/2,-1/2) ; gamma_sin: offdiag -i/2 (0,1), +i/2 (1,0)
    if (t < 4 * NBASIS) {
        int w = t / NBASIS, m = t % NBASIS;
        int md[4] = { m / 27, (m / 9) % 3, (m / 3) % 3, m % 3 };
        float cre = 0.f;
        for (int i = 0; i < 16; ++i) {
            for (int j = 0; j < 16; ++j) {
                float pr = 1.f, pi = 0.f;
                for (int u = 0; u < 4; ++u) {
                    int bi = (i >> (3 - u)) & 1, bj = (j >> (3 - u)) & 1;
                    float gr = 0.f, gi = 0.f;
                    int mu = md[u];
                    if (mu == 0)      { gr = (bi == bj) ? 0.5f : 0.f; }
                    else if (mu == 1) { gr = (bi == bj) ? (bi ? -0.5f : 0.5f) : 0.f; }
                    else              { gi = (bi == bj) ? 0.f : (bi ? 0.5f : -0.5f); }
                    float nr = pr * gr - pi * gi;
                    float ni = pr * gi + pi * gr;
                    pr = nr; pi = ni;
                }
                cre += Mre[w][(i << 4) | j] * pr - Mim[w][(i << 4) | j] * pi;
            }
        }
        Cmat[m * 16 + w] = cre;
    }
}

// ------------------------------------------------------------------
// Main: one wave32 per 16-patch tile.
// A (16 patches x 84)  x  B (84 x 16, cols 0..3 = Cmat)  via 21 chained
// V_WMMA_F32_16X16X4_F32; D tile columns 0..3 are the 4 PauliZ expvals.
// ------------------------------------------------------------------
__global__ __launch_bounds__(256) void quanv_main(const float* __restrict__ x,
                                                  const float* __restrict__ Cmat,
                                                  float* __restrict__ out)
{
    const int lane = threadIdx.x & 31;
    const int wave = threadIdx.x >> 5;
    const int tile = blockIdx.x * 8 + wave;       // 16 patches / tile
    const int pl   = lane & 15;                   // A-row (patch) for this lane (both halves)
    const int hi   = lane >> 4;                   // K sub-offset selector per f32 WMMA A/B layout

    // patch -> (b, oh, ow); 128x128 patches per image, B*16384 patches total
    const int gp  = tile * 16 + pl;
    const int b   = gp >> 14;
    const int p2  = gp & 16383;
    const int ohi = p2 >> 7;
    const int owi = p2 & 127;

    // load the 2x2 patch: two float2 rows (W = 256)
    const int row0 = ((b << 8) + (ohi << 1)) * 256 + (owi << 1);
    v2f a01 = *reinterpret_cast<const v2f*>(x + row0);
    v2f a23 = *reinterpret_cast<const v2f*>(x + row0 + 256);

    float cth[4], sth[4];
    cth[0] = __cosf(a01.x); sth[0] = __sinf(a01.x);
    cth[1] = __cosf(a01.y); sth[1] = __sinf(a01.y);
    cth[2] = __cosf(a23.x); sth[2] = __sinf(a23.x);
    cth[3] = __cosf(a23.y); sth[3] = __sinf(a23.y);

    // product tree for the 81 basis functions
    float g01[9], g23[9];
    {
        float f0[3] = {1.f, cth[0], sth[0]};
        float f1[3] = {1.f, cth[1], sth[1]};
        float f2[3] = {1.f, cth[2], sth[2]};
        float f3[3] = {1.f, cth[3], sth[3]};
#pragma unroll
        for (int i = 0; i < 3; ++i)
#pragma unroll
            for (int j = 0; j < 3; ++j) {
                g01[3 * i + j] = f0[i] * f1[j];
                g23[3 * i + j] = f2[i] * f3[j];
            }
    }

    // B operand: coefficient matrix, invariant -> keep in registers for whole kernel
    const int col  = pl;               // N column
    const int koff = hi ? 2 : 0;       // f32 WMMA: VGPR0 holds K=0 (lanes 0-15) / K=2 (lanes 16-31)
    v2f bmat[NCHUNK];
#pragma unroll
    for (int q = 0; q < NCHUNK; ++q) {
        int k0 = 4 * q + koff;
        bmat[q].x = Cmat[k0 * 16 + col];
        bmat[q].y = Cmat[(k0 + 1) * 16 + col];
    }

    v8f acc = {0.f, 0.f, 0.f, 0.f, 0.f, 0.f, 0.f, 0.f};
#pragma unroll
    for (int q = 0; q < NCHUNK; ++q) {
        float phi[4];
#pragma unroll
        for (int r = 0; r < 4; ++r) {
            int k = 4 * q + r;   // constant after unroll -> g01/g23 stay in registers
            phi[r] = (k < NBASIS)
                   ? g01[(k / 27) * 3 + (k / 9) % 3] * g23[((k / 3) % 3) * 3 + (k % 3)]
                   : 0.f;
        }
        v2f amat;
        amat.x = hi ? phi[2] : phi[0];
        amat.y = hi ? phi[3] : phi[1];
        acc = __builtin_amdgcn_wmma_f32_16x16x4_f32(false, amat, false, bmat[q],
                                                    (short)0, acc, false, false);
    }

    // D layout: VGPR v, lanes 0-15 -> patch v, lanes 16-31 -> patch v+8; N = lane&15.
    // Columns 0..3 hold the 4 expvals; out shape (B, 4, 128, 128).
    if (pl < 4) {
        const int w = pl;
#pragma unroll
        for (int v = 0; v < 8; ++v) {
            int g  = tile * 16 + v + 8 * hi;
            int b2 = g >> 14;
            int q2 = g & 16383;
            out[(((b2 << 2) + w) << 14) + q2] = acc[v];
        }
    }
}

extern "C" void kernel_launch(void* const* d_in, const int* in_sizes, int n_in,
                              void* d_out, int out_size, void* d_ws, size_t ws_size,
                              hipStream_t stream)
{
    const float* x       = (const float*)d_in[0];   // (B,1,256,256)
    const float* weights = (const float*)d_in[1];   // (4,3)
    float*       out     = (float*)d_out;           // (B,4,128,128)
    float*       Cmat    = (float*)d_ws;            // KPAD*16 floats

    const int total   = in_sizes[0];                // B*256*256
    const int B       = total / (256 * 256);
    const int tiles   = (B * 128 * 128) / 16;       // 16 patches per wave
    const int blocks  = tiles / 8;                  // 8 waves per 256-thread block

    quanv_setup<<<1, 512, 0, stream>>>(weights, Cmat);
    quanv_main<<<blocks, 256, 0, stream>>>(x, Cmat, out);
}